// ModelNew_78847009620052
// MI455X (gfx1250) — compile-verified
//
#include <hip/hip_runtime.h>
#include <hip/hip_bf16.h>

// Problem constants (match reference)
#define B_DIM   4096
#define IN_DIM  8192
#define HID_DIM 8192
#define ROWCHUNKS 64                      // stage-1 partial row chunks
#define ROWS_PER_CHUNK (HID_DIM / ROWCHUNKS)   // 128

typedef __attribute__((ext_vector_type(2))) float v2f;
typedef __attribute__((ext_vector_type(4))) float f4;
typedef __attribute__((ext_vector_type(8))) float v8f;

// ---------------------------------------------------------------------------
// Stage 1: partial column sums of W, streamed through LDS with CDNA5 async
// global->LDS loads (ASYNCcnt double-buffer pipeline, GV addressing mode).
// Each block: 256 threads x 4 consecutive columns (16B/lane) over 128 rows.
// grid = (IN/1024 colBlocks, 64 rowChunks). W is read exactly once (256 MB).
// ---------------------------------------------------------------------------
__global__ __launch_bounds__(256)
void colsum_stage1_async(const float* __restrict__ W, float* __restrict__ part) {
    __shared__ __align__(16) float buf[256 * 8];   // 2 x 16B per thread = 8 KB
    const int tid = threadIdx.x;
    const int col = blockIdx.x * 1024 + tid * 4;
    const int row0 = blockIdx.y * ROWS_PER_CHUNK;

    const float* g = W + (size_t)row0 * IN_DIM + col;
    // Generic pointer to __shared__ : low 32 bits are the LDS byte offset.
    const unsigned lds0 = (unsigned)(size_t)(&buf[tid * 8]);
    const unsigned lds1 = lds0 + 16u;

    f4 acc = (f4)(0.0f);

    // Prologue: fill buffer 0 for row 0.
    asm volatile("global_load_async_to_lds_b128 %0, %1, off"
                 :: "v"(lds0), "v"(g) : "memory");

    for (int r = 0; r < ROWS_PER_CHUNK; ++r) {
        const float* gn = g + IN_DIM;
        if (r + 1 < ROWS_PER_CHUNK) {
            // Issue next row into the other buffer, then wait until only that
            // newest op is outstanding (async loads complete in order).
            asm volatile("global_load_async_to_lds_b128 %0, %1, off"
                         :: "v"((r & 1) ? lds0 : lds1), "v"(gn) : "memory");
            asm volatile("s_wait_asynccnt 0x1" ::: "memory");
        } else {
            asm volatile("s_wait_asynccnt 0x0" ::: "memory");
        }
        const f4 v = *(const f4*)&buf[tid * 8 + ((r & 1) ? 4 : 0)];
        acc += v;
        g = gn;
    }
    *(f4*)(part + (size_t)blockIdx.y * IN_DIM + col) = acc;
}

// ---------------------------------------------------------------------------
// Stage 2: fold the 64 partial rows into w_sum using V_WMMA_F32_16X16X4_F32
// with A = all-ones: D = 1(16x4) x B(4x16) + C  =>  each WMMA adds 4 partial
// rows across 16 columns, exact in f32. One wave handles 16 columns.
// B layout (32-bit, 4x16): v0 lanes0-15=K0, lanes16-31=K1; v1: K2, K3.
// D layout: VGPR0 lanes0-15 = row M=0, N=lane  ->  the column sums.
// ---------------------------------------------------------------------------
__global__ __launch_bounds__(256)
void colsum_stage2_wmma(const float* __restrict__ part, float* __restrict__ wsum) {
    const int wid  = threadIdx.x >> 5;
    const int lane = threadIdx.x & 31;
    const int c0   = (blockIdx.x * (blockDim.x >> 5) + wid) * 16;
#if __has_builtin(__builtin_amdgcn_wmma_f32_16x16x4_f32)
    const int colIdx = c0 + (lane & 15);
    const int rowSel = lane >> 4;
    v8f acc = {};
    const v2f a = {1.0f, 1.0f};   // all-ones A matrix
    for (int kb = 0; kb < ROWCHUNKS; kb += 4) {
        v2f b;
        b.x = part[(size_t)(kb + rowSel)     * IN_DIM + colIdx];
        b.y = part[(size_t)(kb + 2 + rowSel) * IN_DIM + colIdx];
        acc = __builtin_amdgcn_wmma_f32_16x16x4_f32(
                  false, a, false, b, (short)0, acc, false, false);
    }
    if (lane < 16) wsum[c0 + lane] = acc[0];
#else
    if (lane < 16) {
        float s = 0.0f;
        for (int r = 0; r < ROWCHUNKS; ++r)
            s += part[(size_t)r * IN_DIM + c0 + lane];
        wsum[c0 + lane] = s;
    }
#endif
}

// ---------------------------------------------------------------------------
// Stage 3: out[b] = 0.75f * dot(x[b], w_sum).  One block per batch row.
// x is read once (128 MB) with nontemporal 128-bit loads; w_sum (32 KB) is
// hot in L2. wave32 shuffle reduction + tiny LDS cross-wave combine.
// ---------------------------------------------------------------------------
__global__ __launch_bounds__(256)
void rowdot_scale(const float* __restrict__ x, const float* __restrict__ wsum,
                  float* __restrict__ out) {
    __shared__ float red[8];
    const int b = blockIdx.x;
    const f4* __restrict__ xr = (const f4*)(x + (size_t)b * IN_DIM);
    const f4* __restrict__ wv = (const f4*)wsum;

    float s = 0.0f;
    #pragma unroll 4
    for (int i = threadIdx.x; i < IN_DIM / 4; i += 256) {
        const f4 xv = __builtin_nontemporal_load(&xr[i]);   // stream-once
        const f4 ws = wv[i];                                 // reused, cached
        s += xv.x * ws.x + xv.y * ws.y + xv.z * ws.z + xv.w * ws.w;
    }
    // wave32 reduction
    for (int off = 16; off > 0; off >>= 1)
        s += __shfl_down(s, off, 32);
    if ((threadIdx.x & 31) == 0) red[threadIdx.x >> 5] = s;
    __syncthreads();
    if (threadIdx.x == 0) {
        float t = 0.0f;
        #pragma unroll
        for (int i = 0; i < 8; ++i) t += red[i];
        out[b] = 0.75f * t;      // (h/2) summed, * 1.5
    }
}

// ---------------------------------------------------------------------------
extern "C" void kernel_launch(void* const* d_in, const int* in_sizes, int n_in,
                              void* d_out, int out_size, void* d_ws, size_t ws_size,
                              hipStream_t stream) {
    const float* x = (const float*)d_in[0];      // [B, IN] f32
    const float* W = (const float*)d_in[1];      // [HID, IN] f32
    float* out = (float*)d_out;                  // [B, 1] f32

    float* part = (float*)d_ws;                          // [ROWCHUNKS, IN]
    float* wsum = part + (size_t)ROWCHUNKS * IN_DIM;     // [IN]

    // Stage 1: W column partial sums (256 MB stream, async->LDS pipeline)
    dim3 g1(IN_DIM / 1024, ROWCHUNKS);
    colsum_stage1_async<<<g1, 256, 0, stream>>>(W, part);

    // Stage 2: fold partials with f32 WMMA (16 cols per wave, 8 waves/block)
    dim3 g2(IN_DIM / (16 * 8));
    colsum_stage2_wmma<<<g2, 256, 0, stream>>>(part, wsum);

    // Stage 3: per-row dot + scale (128 MB stream)
    rowdot_scale<<<B_DIM, 256, 0, stream>>>(x, wsum, out);
}